// ULayer_39900246180085
// MI455X (gfx1250) — compile-verified
//
#include <hip/hip_runtime.h>

// CDNA5 / gfx1250. wave32. Single-pass fused Kronecker apply:
//   out = (U0 (x) U1 (x) U2) @ X,  U_k = cos(t_k) I + sin(t_k) S  (16x16)
// Axis-0 handled as a register Givens rotation fused with the global load;
// axes 1 and 2 handled as 16x16 f32 GEMMs on v_wmma_f32_16x16x4_f32 in LDS.

typedef __attribute__((ext_vector_type(2))) float v2f;
typedef __attribute__((ext_vector_type(8))) float v8f;

#define LDS_STRIDE 33            // 32 batch cols + 1 pad float (bank spread)
#define BLK_FLOATS (256 * LDS_STRIDE)

// U[m,j] for U = c*I + s*S with S[i,15-i]=+1 (i<8), S[i,15-i]=-1 (i>=8)
__device__ __forceinline__ float gate_coef(int m, int j, float c, float s) {
  float v = 0.0f;
  if (j == m) v = c;
  else if (j == 15 - m) v = (m < 8) ? s : -s;
  return v;
}

__global__ __launch_bounds__(256) void ulayer_kron_kernel(
    const float* __restrict__ x,      // [4096, 2048]
    const float* __restrict__ thetas, // [3]
    float* __restrict__ out)          // [4096, 2048]
{
  __shared__ float lds[2 * BLK_FLOATS];

  const int i0  = blockIdx.x;        // 0..7  -> row blocks i0 and 15-i0
  const int bt  = blockIdx.y * 32;   // batch column base
  const int tid = threadIdx.x;

  const float t0 = thetas[0], t1 = thetas[1], t2 = thetas[2];
  const float c0 = cosf(t0), s0 = sinf(t0);
  const float c1 = cosf(t1), s1 = sinf(t1);
  const float c2 = cosf(t2), s2 = sinf(t2);

  const int rowLo = i0 * 256;        // output digit i0 (< 8): +s partner
  const int rowHi = (15 - i0) * 256; // output digit 15-i0 (>=8): -s partner

  // ---- Phase 1: coalesced load + axis-0 rotation + stage into LDS --------
#pragma unroll
  for (int it = 0; it < 8; ++it) {
    const int flat = tid + it * 256;     // 0..2047
    const int q = flat & 7;              // float4 index within 32 cols
    const int r = flat >> 3;             // row within 256-row block
    const int col = bt + q * 4;
    const float4 xlo = *(const float4*)(x + (rowLo + r) * 2048 + col);
    const float4 xhi = *(const float4*)(x + (rowHi + r) * 2048 + col);
    float* plo = &lds[0 * BLK_FLOATS + r * LDS_STRIDE + q * 4];
    float* phi = &lds[1 * BLK_FLOATS + r * LDS_STRIDE + q * 4];
    plo[0] = c0 * xlo.x + s0 * xhi.x;  phi[0] = c0 * xhi.x - s0 * xlo.x;
    plo[1] = c0 * xlo.y + s0 * xhi.y;  phi[1] = c0 * xhi.y - s0 * xlo.y;
    plo[2] = c0 * xlo.z + s0 * xhi.z;  phi[2] = c0 * xhi.z - s0 * xlo.z;
    plo[3] = c0 * xlo.w + s0 * xhi.w;  phi[3] = c0 * xhi.w - s0 * xlo.w;
  }
  __syncthreads();

  const int lane = tid & 31;
  const int half = lane >> 4;   // 0: lanes 0-15, 1: lanes 16-31
  const int l16  = lane & 15;
  const int wave = tid >> 5;    // 8 waves

  // One in-LDS stage: 64 tiles of D = U*T (16x16), 8 tiles per wave,
  // each tile = 4 chained v_wmma_f32_16x16x4_f32.
  auto wmma_stage = [&](float c, float s, int tileRowMul, int kRowStride) {
#pragma unroll
    for (int j = 0; j < 8; ++j) {
      const int tile = wave + (j << 3);       // 0..63, disjoint per wave
      const int qb   = tile >> 5;             // which 256-row block
      const int rem  = tile & 31;
      const int t    = rem >> 1;              // tile index along rows
      const int h    = rem & 1;               // 16-col half of the 32 cols
      float* base = &lds[qb * BLK_FLOATS + t * tileRowMul + h * 16 + l16];

      v8f acc = {};
#pragma unroll
      for (int k = 0; k < 4; ++k) {
        const int k0 = 4 * k + half * 2;      // f32 operand layout: VGPR v ->
        v2f a, b;                             //  K = k0+v (low/high lane split)
        a.x = gate_coef(l16, k0,     c, s);
        a.y = gate_coef(l16, k0 + 1, c, s);
        b.x = base[(k0)     * kRowStride];
        b.y = base[(k0 + 1) * kRowStride];
        acc = __builtin_amdgcn_wmma_f32_16x16x4_f32(
            /*neg_a=*/false, a, /*neg_b=*/false, b,
            /*c_mod=*/(short)0, acc, /*reuse_a=*/false, /*reuse_b=*/false);
      }
      // D layout: VGPR jj -> M = jj + half*8, N = l16 ; write back in place.
#pragma unroll
      for (int jj = 0; jj < 8; ++jj)
        base[(half * 8 + jj) * kRowStride] = acc[jj];
    }
  };

  // ---- Phase 2: axis-2 (16 consecutive rows), U2 -------------------------
  wmma_stage(c2, s2, /*tileRowMul=*/16 * LDS_STRIDE, /*kRowStride=*/LDS_STRIDE);
  __syncthreads();

  // ---- Phase 3: axis-1 (stride-16 rows), U1 ------------------------------
  wmma_stage(c1, s1, /*tileRowMul=*/LDS_STRIDE, /*kRowStride=*/16 * LDS_STRIDE);
  __syncthreads();

  // ---- Phase 4: coalesced store ------------------------------------------
#pragma unroll
  for (int it = 0; it < 8; ++it) {
    const int flat = tid + it * 256;
    const int q = flat & 7;
    const int r = flat >> 3;
    const int col = bt + q * 4;
    const float* plo = &lds[0 * BLK_FLOATS + r * LDS_STRIDE + q * 4];
    const float* phi = &lds[1 * BLK_FLOATS + r * LDS_STRIDE + q * 4];
    float4 vlo, vhi;
    vlo.x = plo[0]; vlo.y = plo[1]; vlo.z = plo[2]; vlo.w = plo[3];
    vhi.x = phi[0]; vhi.y = phi[1]; vhi.z = phi[2]; vhi.w = phi[3];
    *(float4*)(out + (rowLo + r) * 2048 + col) = vlo;
    *(float4*)(out + (rowHi + r) * 2048 + col) = vhi;
  }
}

extern "C" void kernel_launch(void* const* d_in, const int* in_sizes, int n_in,
                              void* d_out, int out_size, void* d_ws, size_t ws_size,
                              hipStream_t stream) {
  (void)in_sizes; (void)n_in; (void)out_size; (void)d_ws; (void)ws_size;
  const float* x      = (const float*)d_in[0];  // [4096*2048] f32
  const float* thetas = (const float*)d_in[1];  // [3] f32
  float* out          = (float*)d_out;          // [4096*2048] f32

  dim3 grid(8, 64, 1);   // 8 axis-0 pair-blocks x 64 batch tiles of 32 cols
  dim3 block(256, 1, 1); // 8 waves (wave32)
  ulayer_kron_kernel<<<grid, block, 0, stream>>>(x, thetas, out);
}